// EncodeArray_25185688224041
// MI455X (gfx1250) — compile-verified
//
#include <hip/hip_runtime.h>
#include <hip/hip_bf16.h>

#define N_NODES 5120
#define N_EDGES 97280
#define E2_TOT  (N_EDGES + N_NODES)   // with self loops
#define NHEADS  4

typedef float v2f __attribute__((ext_vector_type(2)));
typedef float v8f __attribute__((ext_vector_type(8)));

#if defined(__gfx1250__) && __has_builtin(__builtin_amdgcn_wmma_f32_16x16x4_f32)
#define HAVE_WMMA_F32 1
#else
#define HAVE_WMMA_F32 0
#endif

// ---------- helpers ----------
__device__ __forceinline__ float atomicMaxF(float* addr, float v) {
  // classic bit trick: valid for mixed-sign floats
  if (v >= 0.f) return __int_as_float(atomicMax((int*)addr, __float_as_int(v)));
  return __uint_as_float(atomicMin((unsigned int*)addr, __float_as_uint(v)));
}

__global__ void fill_kernel(float* __restrict__ p, float v, long long n) {
  for (long long i = (long long)blockIdx.x * blockDim.x + threadIdx.x; i < n;
       i += (long long)gridDim.x * blockDim.x)
    p[i] = v;
}

// ---------- graph preprocessing ----------
__global__ void deg_kernel(const int* __restrict__ ei, const float* __restrict__ ea,
                           float* __restrict__ deg, float* __restrict__ lsum) {
  for (int e = blockIdx.x * blockDim.x + threadIdx.x; e < N_EDGES;
       e += gridDim.x * blockDim.x) {
    int d = ei[N_EDGES + e];
    atomicAdd(&deg[d], 1.f);
    atomicAdd(&lsum[d], ea[e]);
  }
}

__global__ void loop_attr_kernel(const float* __restrict__ deg,
                                 const float* __restrict__ lsum,
                                 float* __restrict__ lattr) {
  int n = blockIdx.x * blockDim.x + threadIdx.x;
  if (n < N_NODES) lattr[n] = lsum[n] / fmaxf(deg[n], 1.f);
}

// ---------- per-block node linear (input dim = 2: GEMV, scalar is exact) ----------
__global__ void node_lin_kernel(const float* __restrict__ x,
                                const float* __restrict__ Wl, const float* __restrict__ bl,
                                const float* __restrict__ Wr, const float* __restrict__ br,
                                float* __restrict__ xl, float* __restrict__ xr, int hc) {
  long long tot = (long long)N_NODES * hc;
  for (long long t = (long long)blockIdx.x * blockDim.x + threadIdx.x; t < tot;
       t += (long long)gridDim.x * blockDim.x) {
    int n = (int)(t / hc), j = (int)(t % hc);
    float x0 = x[2 * n], x1 = x[2 * n + 1];
    xl[t] = Wl[2 * j] * x0 + Wl[2 * j + 1] * x1 + bl[j];
    xr[t] = Wr[2 * j] * x0 + Wr[2 * j + 1] * x1 + br[j];
  }
}

// ---------- edge logits + segment max (one wave32 per edge, coalesced) ----------
__global__ void edge_logits_kernel(const int* __restrict__ ei, const float* __restrict__ ea,
                                   const float* __restrict__ lattr,
                                   const float* __restrict__ xl, const float* __restrict__ xr,
                                   const float* __restrict__ We, const float* __restrict__ att,
                                   float* __restrict__ logits, float* __restrict__ smax, int C) {
  const int H = NHEADS, hc = H * C;
  const int lane = threadIdx.x & 31;
  const int gw   = (blockIdx.x * blockDim.x + threadIdx.x) >> 5;
  const int nw   = (gridDim.x * blockDim.x) >> 5;
  for (int e = gw; e < E2_TOT; e += nw) {
    int s, d; float eav;
    if (e < N_EDGES) { s = ei[e]; d = ei[N_EDGES + e]; eav = ea[e]; }
    else             { s = d = e - N_EDGES; eav = lattr[s]; }
    const float* xls = xl + (size_t)s * hc;
    const float* xrd = xr + (size_t)d * hc;
    for (int h = 0; h < H; ++h) {
      float acc = 0.f;
      const int base = h * C;
      for (int c = lane; c < C; c += 32) {
        float v = xls[base + c] + xrd[base + c] + eav * We[base + c];
        v = v > 0.f ? v : 0.2f * v;             // leaky_relu(0.2)
        acc += att[base + c] * v;
      }
      // wave32 butterfly reduction
      for (int off = 16; off > 0; off >>= 1) acc += __shfl_xor(acc, off, 32);
      if (lane == 0) {
        logits[(size_t)e * H + h] = acc;
        atomicMaxF(&smax[(size_t)d * H + h], acc);
      }
    }
  }
}

// ---------- exp + segment sum ----------
__global__ void edge_exp_kernel(const int* __restrict__ ei,
                                float* __restrict__ logits,
                                const float* __restrict__ smax,
                                float* __restrict__ den) {
  const int H = NHEADS;
  long long tot = (long long)E2_TOT * H;
  for (long long t = (long long)blockIdx.x * blockDim.x + threadIdx.x; t < tot;
       t += (long long)gridDim.x * blockDim.x) {
    int e = (int)(t / H), h = (int)(t % H);
    int d = (e < N_EDGES) ? ei[N_EDGES + e] : (e - N_EDGES);
    float ex = __expf(logits[t] - smax[(size_t)d * H + h]);
    logits[t] = ex;
    atomicAdd(&den[(size_t)d * H + h], ex);
  }
}

// ---------- alpha-weighted aggregation (one wave32 per edge, coalesced) ----------
__global__ void edge_agg_kernel(const int* __restrict__ ei,
                                const float* __restrict__ xl,
                                const float* __restrict__ exv,
                                const float* __restrict__ den,
                                float* __restrict__ agg, int C) {
  const int H = NHEADS, hc = H * C;
  const int lane = threadIdx.x & 31;
  const int gw   = (blockIdx.x * blockDim.x + threadIdx.x) >> 5;
  const int nw   = (gridDim.x * blockDim.x) >> 5;
  for (int e = gw; e < E2_TOT; e += nw) {
    int s, d;
    if (e < N_EDGES) { s = ei[e]; d = ei[N_EDGES + e]; }
    else             { s = d = e - N_EDGES; }
    const float* xls = xl + (size_t)s * hc;
    float* aggd = agg + (size_t)d * hc;
    for (int j = lane; j < hc; j += 32) {
      int h = j / C;
      float alpha = exv[(size_t)e * H + h] / den[(size_t)d * H + h];
      atomicAdd(&aggd[j], xls[j] * alpha);
    }
  }
}

// ---------- head mean + bias + relu + projected residual ----------
__global__ void combine_kernel(const float* __restrict__ x,
                               const float* __restrict__ agg,
                               const float* __restrict__ bias,
                               const float* __restrict__ Wres,
                               float* __restrict__ xb, int C) {
  const int H = NHEADS, hc = H * C;
  long long tot = (long long)N_NODES * C;
  for (long long t = (long long)blockIdx.x * blockDim.x + threadIdx.x; t < tot;
       t += (long long)gridDim.x * blockDim.x) {
    int n = (int)(t / C), c = (int)(t % C);
    float g = 0.f;
    for (int h = 0; h < H; ++h) g += agg[(size_t)n * hc + h * C + c];
    g = g * 0.25f + bias[c];
    float r = g > 0.f ? g : 0.f;
    xb[t] = r + Wres[2 * c] * x[2 * n] + Wres[2 * c + 1] * x[2 * n + 1];
  }
}

// ---------- fp32 WMMA GEMM with LDS tiling ----------
// C[m, colOff+n] = (relu?)(A[m,:] dot W[n,:] + bias[n]);  W row-major [Nc, K] (B^T).
// Block = 128 threads (4 waves). Tile: BM=16, BN=128, KC=32.
// Each wave owns two 16x16 C tiles (reuses its A fragment across both WMMAs).
// A chunk (16x32, 2KB) and B chunk (128x32, 17KB padded) staged in LDS with
// coalesced global loads; WMMA operands come from ds_load.
// Fragment layouts per ISA 7.12.2:
//   A 16x4 f32: lane=16*half+l -> M=l, K=k+2*half+{0,1}
//   B 4x16 f32: rows k+half, k+2+half; N=l
//   C/D:        VGPR i -> M=i+8*half, N=l
#define KC   32
#define LDSP (KC + 2)   // pad to spread LDS banks, keeps 8B alignment for b64 reads
__global__ __launch_bounds__(128)
void gemm_wmma_kernel(const float* __restrict__ A, int lda,
                      const float* __restrict__ W, int K,
                      const float* __restrict__ bias,
                      float* __restrict__ Cout, int ldc, int colOff, int doRelu) {
  __shared__ float As[16 * LDSP];
  __shared__ float Bs[128 * LDSP];

  const int tid  = threadIdx.x;
  const int lane = tid & 31;
  const int wv   = tid >> 5;
  const int l    = lane & 15;
  const int half = lane >> 4;
  const int tm   = blockIdx.x * 16;
  const int tnB  = blockIdx.y * 128;            // block's first output column
  const int n0   = wv * 32;                     // wave's first local column
  const int n1   = n0 + 16;

  // cooperative-load index precompute
  const int ar = tid >> 3;                      // A: row 0..15
  const int ac = (tid & 7) * 4;                 // A: col group of 4

  v8f acc0 = {0.f, 0.f, 0.f, 0.f, 0.f, 0.f, 0.f, 0.f};
  v8f acc1 = {0.f, 0.f, 0.f, 0.f, 0.f, 0.f, 0.f, 0.f};

  for (int k0 = 0; k0 < K; k0 += KC) {
    // ---- stage A (16x32): 128 thr x 4 floats, coalesced ----
    {
      const float* src = A + (size_t)(tm + ar) * lda + k0 + ac;
      __builtin_prefetch(src + KC, 0, 0);       // speculative prefetch of next chunk
#pragma unroll
      for (int i = 0; i < 4; ++i) As[ar * LDSP + ac + i] = src[i];
    }
    // ---- stage B (128x32): thread t loads row t (contiguous in K) ----
    {
      const float* src = W + (size_t)(tnB + tid) * K + k0;
      __builtin_prefetch(src + KC, 0, 0);
#pragma unroll
      for (int i = 0; i < KC; i += 4) {
        Bs[tid * LDSP + i + 0] = src[i + 0];
        Bs[tid * LDSP + i + 1] = src[i + 1];
        Bs[tid * LDSP + i + 2] = src[i + 2];
        Bs[tid * LDSP + i + 3] = src[i + 3];
      }
    }
    __syncthreads();

#if HAVE_WMMA_F32
#pragma unroll
    for (int kk = 0; kk < KC; kk += 4) {
      v2f a, b0, b1;
      a.x  = As[l * LDSP + kk + 2 * half];
      a.y  = As[l * LDSP + kk + 2 * half + 1];
      b0.x = Bs[(n0 + l) * LDSP + kk + half];
      b0.y = Bs[(n0 + l) * LDSP + kk + 2 + half];
      b1.x = Bs[(n1 + l) * LDSP + kk + half];
      b1.y = Bs[(n1 + l) * LDSP + kk + 2 + half];
      acc0 = __builtin_amdgcn_wmma_f32_16x16x4_f32(false, a, false, b0,
                                                   (short)0, acc0, false, false);
      acc1 = __builtin_amdgcn_wmma_f32_16x16x4_f32(false, a, false, b1,
                                                   (short)0, acc1, false, false);
    }
#else
    // scalar fallback (host-side parse / missing builtin)
    for (int kk = 0; kk < KC; ++kk) {
#pragma unroll
      for (int i = 0; i < 8; ++i) {
        float av = As[(i + 8 * half) * LDSP + kk];
        acc0[i] += av * Bs[(n0 + l) * LDSP + kk];
        acc1[i] += av * Bs[(n1 + l) * LDSP + kk];
      }
    }
#endif
    __syncthreads();
  }

#pragma unroll
  for (int i = 0; i < 8; ++i) {
    int row  = tm + i + 8 * half;
    int col0 = tnB + n0 + l;
    int col1 = tnB + n1 + l;
    float v0 = acc0[i] + bias[col0];
    float v1 = acc1[i] + bias[col1];
    if (doRelu) { v0 = fmaxf(v0, 0.f); v1 = fmaxf(v1, 0.f); }
    Cout[(size_t)row * ldc + colOff + col0] = v0;
    Cout[(size_t)row * ldc + colOff + col1] = v1;
  }
}

// ---------- launch ----------
static inline dim3 nb(long long n, int t) {
  long long b = (n + t - 1) / t;
  if (b > 262144) b = 262144;
  return dim3((unsigned)b);
}

extern "C" void kernel_launch(void* const* d_in, const int* in_sizes, int n_in,
                              void* d_out, int out_size, void* d_ws, size_t ws_size,
                              hipStream_t stream) {
  (void)in_sizes; (void)n_in; (void)out_size; (void)ws_size;
  const float* x  = (const float*)d_in[0];
  const int*   ei = (const int*)d_in[1];
  const float* ea = (const float*)d_in[2];
  const float* Wf = (const float*)d_in[33];
  const float* bf = (const float*)d_in[34];

  const int N = N_NODES, E = N_EDGES, E2 = E2_TOT, H = NHEADS;
  const int Cs[3]  = {64, 128, 256};
  const int hcs[3] = {256, 512, 1024};
  const int hcoff[3] = {0, 256, 768};   // cumulative hc
  const int coff[3]  = {0, 64, 192};    // cumulative C
  const int HCSUM = 1792, CSUM = 448;

  float* ws = (float*)d_ws;
  size_t off = 0;
  float* deg   = ws + off; off += N;
  float* lsum  = ws + off; off += N;
  float* lattr = ws + off; off += N;
  float* xl    = ws + off; off += (size_t)N * HCSUM;
  float* xr    = ws + off; off += (size_t)N * HCSUM;
  float* agg   = ws + off; off += (size_t)N * HCSUM;
  float* exb   = ws + off; off += (size_t)E2 * H * 3;
  float* smax  = ws + off; off += (size_t)N * H * 3;
  float* sden  = ws + off; off += (size_t)N * H * 3;
  float* xb    = ws + off; off += (size_t)N * CSUM;
  float* xc    = ws + off; off += (size_t)N * 1536;

  // init (deg & lsum are contiguous)
  fill_kernel<<<nb(2LL * N, 256), 256, 0, stream>>>(deg, 0.f, 2LL * N);
  fill_kernel<<<nb((long long)N * HCSUM, 256), 256, 0, stream>>>(agg, 0.f, (long long)N * HCSUM);
  fill_kernel<<<nb((long long)N * H * 3, 256), 256, 0, stream>>>(sden, 0.f, (long long)N * H * 3);
  fill_kernel<<<nb((long long)N * H * 3, 256), 256, 0, stream>>>(smax, -3.0e38f, (long long)N * H * 3);

  deg_kernel<<<nb(E, 256), 256, 0, stream>>>(ei, ea, deg, lsum);
  loop_attr_kernel<<<nb(N, 256), 256, 0, stream>>>(deg, lsum, lattr);

  for (int b = 0; b < 3; ++b) {
    const float* Wl   = (const float*)d_in[3 + b * 10 + 0];
    const float* bl   = (const float*)d_in[3 + b * 10 + 1];
    const float* Wr   = (const float*)d_in[3 + b * 10 + 2];
    const float* br   = (const float*)d_in[3 + b * 10 + 3];
    const float* We   = (const float*)d_in[3 + b * 10 + 4];
    const float* att  = (const float*)d_in[3 + b * 10 + 5];
    const float* bias = (const float*)d_in[3 + b * 10 + 6];
    const float* Wres = (const float*)d_in[3 + b * 10 + 7];
    const float* Wp   = (const float*)d_in[3 + b * 10 + 8];
    const float* bp   = (const float*)d_in[3 + b * 10 + 9];
    const int C = Cs[b], hc = hcs[b];
    float* xl_b   = xl  + (size_t)N * hcoff[b];
    float* xr_b   = xr  + (size_t)N * hcoff[b];
    float* agg_b  = agg + (size_t)N * hcoff[b];
    float* ex_b   = exb + (size_t)E2 * H * b;
    float* smax_b = smax + (size_t)N * H * b;
    float* sden_b = sden + (size_t)N * H * b;
    float* xb_b   = xb  + (size_t)N * coff[b];

    node_lin_kernel<<<nb((long long)N * hc, 256), 256, 0, stream>>>(x, Wl, bl, Wr, br, xl_b, xr_b, hc);
    // one wave32 per edge
    edge_logits_kernel<<<nb((long long)E2 * 32, 256), 256, 0, stream>>>(ei, ea, lattr, xl_b, xr_b, We, att, ex_b, smax_b, C);
    edge_exp_kernel<<<nb((long long)E2 * H, 256), 256, 0, stream>>>(ei, ex_b, smax_b, sden_b);
    edge_agg_kernel<<<nb((long long)E2 * 32, 256), 256, 0, stream>>>(ei, xl_b, ex_b, sden_b, agg_b, C);
    combine_kernel<<<nb((long long)N * C, 256), 256, 0, stream>>>(x, agg_b, bias, Wres, xb_b, C);

    dim3 g(N / 16, 512 / 128);  // per-block projection into concat buffer
    gemm_wmma_kernel<<<g, 128, 0, stream>>>(xb_b, C, Wp, C, bp, xc, 1536, 512 * b, 0);
  }

  dim3 g(N / 16, 512 / 128);    // final [5120x1536]x[1536x512] + bias + relu
  gemm_wmma_kernel<<<g, 128, 0, stream>>>(xc, 1536, Wf, 1536, bf, (float*)d_out, 512, 0, 1);
}